// train_model_65343632441985
// MI455X (gfx1250) — compile-verified
//
#include <hip/hip_runtime.h>
#include <hip/hip_bf16.h>
#include <math.h>

typedef __attribute__((ext_vector_type(2))) float v2f;
typedef __attribute__((ext_vector_type(8))) float v8f;

#define D128 128
#define EPS 1e-5f

// ---------------------------------------------------------------------------
// Generic K=128 GEMM: C[M,N] = alpha * act( bnA(A) @ B )
// Compile-time specialized on <NORM_A, TRANS_B, RELU> so the fully-unrolled
// K-loop is pure load + v_wmma_f32_16x16x4_f32 (32 wmma per 16x16 tile).
// grid = (N/16, M/64), block = 128 threads (4 waves, 1 tile each).
// ---------------------------------------------------------------------------
template<bool NORM_A, bool TRANS_B, bool RELU>
__global__ void gemm_k128(const float* __restrict__ A, long lda,
                          const float* __restrict__ Astats,
                          const float* __restrict__ Bm, long ldb,
                          float* __restrict__ C, long ldc,
                          int M, float alpha)
{
    const int lane = threadIdx.x & 31;
    const int wave = threadIdx.x >> 5;
    const int n0 = blockIdx.x * 16;
    const int m0 = blockIdx.y * 64 + wave * 16;
    if (m0 >= M) return;                 // uniform per-wave: EXEC stays all-ones

    const int half = lane >> 4;          // K sub-half (0: k,k+1 | 1: k+2,k+3)
    const int l    = lane & 15;          // row (A) / col (B) within tile

    // Per-lane base pointers (constant across the K loop).
    const float* __restrict__ Arow = A + (long)(m0 + l) * lda + 2 * half;
    const float* __restrict__ Bb   = TRANS_B
        ? (Bm + (long)(n0 + l) * ldb + 2 * half)        // B[c, k] row-major
        : (Bm + (long)(2 * half) * ldb + (n0 + l));     // B[k, c] row-major

    v8f acc = (v8f)0.0f;

    #pragma unroll
    for (int k0 = 0; k0 < D128; k0 += 4) {
        v2f a = *(const v2f*)(Arow + k0);               // 8B-aligned b64 load
        if (NORM_A) {
            const int ka = k0 + 2 * half;
            a.x = (a.x - Astats[2 * ka])     * Astats[2 * ka + 1];
            a.y = (a.y - Astats[2 * ka + 2]) * Astats[2 * ka + 3];
        }
        v2f b;
        if (TRANS_B) {
            b = *(const v2f*)(Bb + k0);                 // contiguous pair
        } else {
            b.x = Bb[(long)k0 * ldb];
            b.y = Bb[(long)k0 * ldb + ldb];
        }
        acc = __builtin_amdgcn_wmma_f32_16x16x4_f32(
                  false, a, false, b, (short)0, acc, false, false);
    }

    const int crow = m0 + half * 8;
    #pragma unroll
    for (int v = 0; v < 8; ++v) {
        float x = acc[v] * alpha;
        if (RELU) x = fmaxf(x, 0.0f);
        C[(long)(crow + v) * ldc + (n0 + l)] = x;
    }
}

// ---------------------------------------------------------------------------
// Cross-encoder stage-1: z1 = relu( (time_all_cls + freq_all_cls) @ W_cross )
// A rows synthesized from the concat/gather mapping (CLS tokens only); the
// mapped row pointers are fixed per lane, so the K loop is branch-free.
// grid = (128/16, 3B/64), block = 128.
// ---------------------------------------------------------------------------
__global__ void gemm_cross(const float* __restrict__ h_t,
                           const float* __restrict__ h_f,
                           const int* __restrict__ idxT,   // time_neg_idx
                           const int* __restrict__ idxF,   // freq_neg_idx
                           const float* __restrict__ W,
                           float* __restrict__ C,
                           int Bsz, long SL)
{
    const int lane = threadIdx.x & 31;
    const int wave = threadIdx.x >> 5;
    const int n0 = blockIdx.x * 16;
    const int m0 = blockIdx.y * 64 + wave * 16;
    if (m0 >= 3 * Bsz) return;

    const int half = lane >> 4;
    const int l    = lane & 15;
    const int r    = m0 + l;

    const float* __restrict__ tr;
    const float* __restrict__ fr;
    if (r < Bsz)             { tr = h_t + (long)r * SL;
                               fr = h_f + (long)r * SL; }
    else if (r < 2 * Bsz)    { int i = r - Bsz;
                               tr = h_t + (long)i * SL;
                               fr = h_f + (long)idxF[i] * SL; }
    else                     { int i = r - 2 * Bsz;
                               tr = h_t + (long)idxT[i] * SL;
                               fr = h_f + (long)i * SL; }
    tr += 2 * half;
    fr += 2 * half;
    const float* __restrict__ Wb = W + (long)(2 * half) * D128 + (n0 + l);

    v8f acc = (v8f)0.0f;
    #pragma unroll
    for (int k0 = 0; k0 < D128; k0 += 4) {
        v2f at = *(const v2f*)(tr + k0);
        v2f af = *(const v2f*)(fr + k0);
        v2f a, b;
        a.x = at.x + af.x;
        a.y = at.y + af.y;
        b.x = Wb[(long)k0 * D128];
        b.y = Wb[(long)k0 * D128 + D128];
        acc = __builtin_amdgcn_wmma_f32_16x16x4_f32(
                  false, a, false, b, (short)0, acc, false, false);
    }

    const int crow = m0 + half * 8;
    #pragma unroll
    for (int v = 0; v < 8; ++v)
        C[(long)(crow + v) * D128 + (n0 + l)] = fmaxf(acc[v], 0.0f);
}

// ---------------------------------------------------------------------------
// Per-column BatchNorm stats: stats[2c]=mean, stats[2c+1]=1/sqrt(var+eps).
// grid = ncols, block = 256.
// ---------------------------------------------------------------------------
__global__ void colstats(const float* __restrict__ Y, long ld, int M,
                         float* __restrict__ stats)
{
    const int c = blockIdx.x;
    float s = 0.0f, s2 = 0.0f;
    for (int r = threadIdx.x; r < M; r += blockDim.x) {
        float v = Y[(long)r * ld + c];
        s += v; s2 += v * v;
    }
    __shared__ float sh[512];
    sh[threadIdx.x] = s;
    sh[256 + threadIdx.x] = s2;
    __syncthreads();
    for (int o = 128; o > 0; o >>= 1) {
        if ((int)threadIdx.x < o) {
            sh[threadIdx.x]       += sh[threadIdx.x + o];
            sh[256 + threadIdx.x] += sh[256 + threadIdx.x + o];
        }
        __syncthreads();
    }
    if (threadIdx.x == 0) {
        float mean = sh[0] / (float)M;
        float var  = sh[256] / (float)M - mean * mean;
        stats[2 * c]     = mean;
        stats[2 * c + 1] = rsqrtf(var + EPS);
    }
}

// ---------------------------------------------------------------------------
// Fused BatchNorm + row L2 normalization (in place). grid = M, block = 128.
// ---------------------------------------------------------------------------
__global__ void bn_rownorm(float* __restrict__ Y,
                           const float* __restrict__ stats, int M)
{
    const int r = blockIdx.x;
    const int c = threadIdx.x;
    float v = (Y[(long)r * D128 + c] - stats[2 * c]) * stats[2 * c + 1];
    __shared__ float sh[D128];
    __shared__ float rn;
    sh[c] = v * v;
    __syncthreads();
    for (int o = 64; o > 0; o >>= 1) {
        if (c < o) sh[c] += sh[c + o];
        __syncthreads();
    }
    if (c == 0) rn = rsqrtf(fmaxf(sh[0], 1e-30f));
    __syncthreads();
    Y[(long)r * D128 + c] = v * rn;
}

// ---------------------------------------------------------------------------
// Gumbel-argmax categorical sampling per row, diagonal masked.
// (log_softmax is shift-invariant, so raw logits + Gumbel noise is exact.)
// grid = B, block = 256.
// ---------------------------------------------------------------------------
__device__ __forceinline__ unsigned pcg_hash(unsigned x)
{
    x = x * 747796405u + 2891336453u;
    unsigned w = ((x >> ((x >> 28) + 4u)) ^ x) * 277803737u;
    return (w >> 22) ^ w;
}

__global__ void sample_neg(const float* __restrict__ logits, int B,
                           unsigned seed, int* __restrict__ idx)
{
    const int i = blockIdx.x;
    float best = -3.0e38f;
    int bj = (i + 1) % B;
    for (int j = threadIdx.x; j < B; j += blockDim.x) {
        if (j == i) continue;
        unsigned h = pcg_hash(((unsigned)i * 2654435761u) ^
                              ((unsigned)j * 2246822519u) ^ seed);
        float u = fmaxf((float)(h >> 8) * (1.0f / 16777216.0f), 1e-7f);
        float g = -__logf(-__logf(u));
        float v = logits[(long)i * B + j] + g;
        if (v > best) { best = v; bj = j; }
    }
    __shared__ float sv[256];
    __shared__ int   si[256];
    sv[threadIdx.x] = best;
    si[threadIdx.x] = bj;
    __syncthreads();
    for (int o = 128; o > 0; o >>= 1) {
        if ((int)threadIdx.x < o && sv[threadIdx.x + o] > sv[threadIdx.x]) {
            sv[threadIdx.x] = sv[threadIdx.x + o];
            si[threadIdx.x] = si[threadIdx.x + o];
        }
        __syncthreads();
    }
    if (threadIdx.x == 0) idx[i] = si[0];
}

// ---------------------------------------------------------------------------
// Final head: z3[r,{0,1}] = bn(z2[r,:]) @ W_pred   (128 -> 2, trivial FLOPs)
// ---------------------------------------------------------------------------
__global__ void pred2(const float* __restrict__ Z2,
                      const float* __restrict__ stats,
                      const float* __restrict__ Wp,
                      float* __restrict__ Z3, int M)
{
    const int r = blockIdx.x * blockDim.x + threadIdx.x;
    if (r >= M) return;
    float a0 = 0.0f, a1 = 0.0f;
    #pragma unroll 8
    for (int k = 0; k < D128; ++k) {
        float v = (Z2[(long)r * D128 + k] - stats[2 * k]) * stats[2 * k + 1];
        a0 += v * Wp[k * 2 + 0];
        a1 += v * Wp[k * 2 + 1];
    }
    Z3[r * 2 + 0] = a0;
    Z3[r * 2 + 1] = a1;
}

__global__ void bn_apply_out(const float* __restrict__ Z3,
                             const float* __restrict__ stats,
                             float* __restrict__ out, int M)
{
    const int t = blockIdx.x * blockDim.x + threadIdx.x;
    if (t >= 2 * M) return;
    const int c = t & 1;
    out[t] = (Z3[t] - stats[2 * c]) * stats[2 * c + 1];
}

// ---------------------------------------------------------------------------
extern "C" void kernel_launch(void* const* d_in, const int* in_sizes, int n_in,
                              void* d_out, int out_size, void* d_ws, size_t ws_size,
                              hipStream_t stream)
{
    const float* h_t       = (const float*)d_in[0];
    const float* h_f       = (const float*)d_in[1];
    const float* W_Tproj   = (const float*)d_in[2];
    const float* W_Fproj   = (const float*)d_in[3];
    const float* W_Tmom    = (const float*)d_in[4];
    const float* W_Fmom    = (const float*)d_in[5];
    const float* W_Tpred   = (const float*)d_in[6];
    const float* W_Fpred   = (const float*)d_in[7];
    const float* W_cross   = (const float*)d_in[8];
    const float* W_proj    = (const float*)d_in[9];
    const float* W_pred    = (const float*)d_in[10];

    const int  B  = in_sizes[0] / (D128 * D128);   // 2048
    const long SL = (long)D128 * D128;             // CLS row stride (S*L)
    const int  M3 = 3 * B;

    float* out = (float*)d_out;
    float* logits_t = out;                 // [B,B]
    float* logits_f = out + (long)B * B;   // [B,B]
    float* z_out    = out + 2L * B * B;    // [3B,2]

    // ---- workspace carve-up ----
    float* ws  = (float*)d_ws;
    float* yT  = ws;               ws += (long)B * D128;
    float* yF  = ws;               ws += (long)B * D128;
    float* kT  = ws;               ws += (long)B * D128;   // momentum (proj only)
    float* kF  = ws;               ws += (long)B * D128;
    float* qT  = ws;               ws += (long)B * D128;
    float* qF  = ws;               ws += (long)B * D128;
    float* z1  = ws;               ws += (long)M3 * D128;
    float* z2  = ws;               ws += (long)M3 * D128;
    float* z3  = ws;               ws += (long)M3 * 2;
    float* sT1 = ws;               ws += 256;
    float* sF1 = ws;               ws += 256;
    float* sTm = ws;               ws += 256;
    float* sFm = ws;               ws += 256;
    float* sT2 = ws;               ws += 256;
    float* sF2 = ws;               ws += 256;
    float* sZ2 = ws;               ws += 256;
    float* sZ3 = ws;               ws += 8;
    int* idxT  = (int*)ws;         // time_neg_idx [B]
    int* idxF  = idxT + B;         // freq_neg_idx [B]

    const dim3 blk(128);
    const dim3 gW(D128 / 16, B / 64);        // N=128 GEMMs over B rows
    const dim3 gL(B / 16, B / 64);           // logits GEMMs
    const dim3 gC(D128 / 16, M3 / 64);       // cross-stage GEMMs

    // ---- projections (CLS rows, stride S*L) ----
    gemm_k128<false, false, false><<<gW, blk, 0, stream>>>(h_t, SL, nullptr, W_Tproj, D128, yT, D128, B, 1.0f);
    gemm_k128<false, false, false><<<gW, blk, 0, stream>>>(h_f, SL, nullptr, W_Fproj, D128, yF, D128, B, 1.0f);
    gemm_k128<false, false, false><<<gW, blk, 0, stream>>>(h_t, SL, nullptr, W_Tmom,  D128, kT, D128, B, 1.0f);
    gemm_k128<false, false, false><<<gW, blk, 0, stream>>>(h_f, SL, nullptr, W_Fmom,  D128, kF, D128, B, 1.0f);
    colstats<<<D128, 256, 0, stream>>>(yT, D128, B, sT1);
    colstats<<<D128, 256, 0, stream>>>(yF, D128, B, sF1);
    colstats<<<D128, 256, 0, stream>>>(kT, D128, B, sTm);
    colstats<<<D128, 256, 0, stream>>>(kF, D128, B, sFm);

    // ---- predictors (BN fused into A-load) ----
    gemm_k128<true, false, false><<<gW, blk, 0, stream>>>(yT, D128, sT1, W_Tpred, D128, qT, D128, B, 1.0f);
    gemm_k128<true, false, false><<<gW, blk, 0, stream>>>(yF, D128, sF1, W_Fpred, D128, qF, D128, B, 1.0f);
    colstats<<<D128, 256, 0, stream>>>(qT, D128, B, sT2);
    colstats<<<D128, 256, 0, stream>>>(qF, D128, B, sF2);

    // ---- BN + L2 row-normalize q_t, q_f, k_t, k_f ----
    bn_rownorm<<<B, D128, 0, stream>>>(qT, sT2, B);
    bn_rownorm<<<B, D128, 0, stream>>>(qF, sF2, B);
    bn_rownorm<<<B, D128, 0, stream>>>(kT, sTm, B);
    bn_rownorm<<<B, D128, 0, stream>>>(kF, sFm, B);

    // ---- logits = (q @ k^T) / TEMP ----
    const float invT = 1.0f / 0.07f;
    gemm_k128<false, true, false><<<gL, blk, 0, stream>>>(qT, D128, nullptr, kF, D128, logits_t, B, B, invT);
    gemm_k128<false, true, false><<<gL, blk, 0, stream>>>(qF, D128, nullptr, kT, D128, logits_f, B, B, invT);

    // ---- categorical sampling (diag masked) ----
    sample_neg<<<B, 256, 0, stream>>>(logits_f, B, 0x9E3779B9u, idxT); // time_neg_idx
    sample_neg<<<B, 256, 0, stream>>>(logits_t, B, 0x85EBCA6Bu, idxF); // freq_neg_idx

    // ---- cross encoder (CLS only) ----
    gemm_cross<<<gC, blk, 0, stream>>>(h_t, h_f, idxT, idxF, W_cross, z1, B, SL);
    gemm_k128<false, false, false><<<gC, blk, 0, stream>>>(z1, D128, nullptr, W_proj, D128, z2, D128, M3, 1.0f);
    colstats<<<D128, 256, 0, stream>>>(z2, D128, M3, sZ2);
    pred2<<<(M3 + 255) / 256, 256, 0, stream>>>(z2, sZ2, W_pred, z3, M3);
    colstats<<<2, 256, 0, stream>>>(z3, 2, M3, sZ3);
    bn_apply_out<<<(2 * M3 + 255) / 256, 256, 0, stream>>>(z3, sZ3, z_out, M3);
}